// LIFAct_csr_9852654977540
// MI455X (gfx1250) — compile-verified
//
#include <hip/hip_runtime.h>

// LIF neuron scan, T=4 fixed (matches reference lax.scan over leading axis).
//   u = u*DECAY + x_t ; s = (u - V_TH > 0) ; u = s ? 0 : u
// Bandwidth-bound: 411 MB streamed once -> ~17.7 us floor at 23.3 TB/s.
// Strategy: one thread owns one float4 across all 4 timesteps; scan state in
// VGPRs; 128-bit non-temporal loads/stores (working set > 192 MB L2, zero reuse).

typedef float v4f __attribute__((ext_vector_type(4)));

#define LIF_DECAY 0.25f
#define LIF_VTH   1.0f

__global__ __launch_bounds__(256) void lif_scan4_vec(
    const float* __restrict__ x, float* __restrict__ out, long long nvec)
{
    long long i = (long long)blockIdx.x * blockDim.x + threadIdx.x;
    if (i >= nvec) return;

    const v4f* __restrict__ xv = (const v4f*)x;
    v4f* __restrict__ ov = (v4f*)out;

    // Four independent 128-bit NT loads in flight (one per timestep).
    v4f x0 = __builtin_nontemporal_load(xv + i);
    v4f x1 = __builtin_nontemporal_load(xv + i + nvec);
    v4f x2 = __builtin_nontemporal_load(xv + i + 2 * nvec);
    v4f x3 = __builtin_nontemporal_load(xv + i + 3 * nvec);

    v4f s0, s1, s2, s3;
#pragma unroll
    for (int k = 0; k < 4; ++k) {
        // t = 0 : u0 = 0*decay + x0 = x0
        float u  = x0[k];
        bool  f  = (u > LIF_VTH);          // spike(u - V_TH): strict >
        s0[k]    = f ? 1.0f : 0.0f;
        u        = f ? 0.0f : u;           // hard reset
        // t = 1
        u        = __builtin_fmaf(u, LIF_DECAY, x1[k]);
        f        = (u > LIF_VTH);
        s1[k]    = f ? 1.0f : 0.0f;
        u        = f ? 0.0f : u;
        // t = 2
        u        = __builtin_fmaf(u, LIF_DECAY, x2[k]);
        f        = (u > LIF_VTH);
        s2[k]    = f ? 1.0f : 0.0f;
        u        = f ? 0.0f : u;
        // t = 3
        u        = __builtin_fmaf(u, LIF_DECAY, x3[k]);
        s3[k]    = (u > LIF_VTH) ? 1.0f : 0.0f;
    }

    __builtin_nontemporal_store(s0, ov + i);
    __builtin_nontemporal_store(s1, ov + i + nvec);
    __builtin_nontemporal_store(s2, ov + i + 2 * nvec);
    __builtin_nontemporal_store(s3, ov + i + 3 * nvec);
}

// Scalar tail for nper % 4 != 0 (not hit for the reference shape; kept for safety).
__global__ void lif_scan4_tail(
    const float* __restrict__ x, float* __restrict__ out,
    long long nper, long long start)
{
    long long i = start + (long long)blockIdx.x * blockDim.x + threadIdx.x;
    if (i >= nper) return;

    float u = 0.0f;
#pragma unroll
    for (int t = 0; t < 4; ++t) {
        u = __builtin_fmaf(u, LIF_DECAY, x[(long long)t * nper + i]);
        bool f = (u > LIF_VTH);
        out[(long long)t * nper + i] = f ? 1.0f : 0.0f;
        u = f ? 0.0f : u;
    }
}

extern "C" void kernel_launch(void* const* d_in, const int* in_sizes, int n_in,
                              void* d_out, int out_size, void* d_ws, size_t ws_size,
                              hipStream_t stream)
{
    (void)n_in; (void)out_size; (void)d_ws; (void)ws_size;
    const float* x = (const float*)d_in[0];
    float* out = (float*)d_out;

    const long long total = (long long)in_sizes[0];   // T * B * C * H * W
    const long long nper  = total / 4;                // elements per timestep (T = 4)
    const long long nvec  = nper / 4;                 // float4 groups per timestep
    const long long rem   = nper - nvec * 4;

    const int block = 256;                            // 8 wave32 per workgroup
    if (nvec > 0) {
        const unsigned grid = (unsigned)((nvec + block - 1) / block);
        lif_scan4_vec<<<grid, block, 0, stream>>>(x, out, nvec);
    }
    if (rem > 0) {
        lif_scan4_tail<<<1, 64, 0, stream>>>(x, out, nper, nvec * 4);
    }
}